// DenseBlock_6803228197635
// MI455X (gfx1250) — compile-verified
//
#include <hip/hip_runtime.h>
#include <hip/hip_bf16.h>

// ---------------------------------------------------------------------------
// DenseBlock GCN (K=16, 4 layers) for MI455X / gfx1250 (wave32, WMMA).
// FC layers use V_WMMA_F32_16X16X4_F32 (full fp32 precision).
// ---------------------------------------------------------------------------

#define NNODES 200000
#define NEDGES 3200000
#define BN_EPS 1e-5f

typedef float v2f __attribute__((ext_vector_type(2)));
typedef float v8f __attribute__((ext_vector_type(8)));

// ---------------------------------------------------------------- utilities
__global__ void zero_f32_kernel(float* __restrict__ p, long n) {
  long i = (long)blockIdx.x * blockDim.x + threadIdx.x;
  long stride = (long)gridDim.x * blockDim.x;
  for (; i < n; i += stride) p[i] = 0.0f;
}

// deg_in[n] = indegree, deg_out[n] = outdegree (self-loop +1 added later)
__global__ void deg_kernel(const int* __restrict__ src, const int* __restrict__ dst,
                           float* __restrict__ deg_in, float* __restrict__ deg_out, long E) {
  long i = (long)blockIdx.x * blockDim.x + threadIdx.x;
  long stride = (long)gridDim.x * blockDim.x;
  for (; i < E; i += stride) {
    atomicAdd(&deg_in[dst[i]], 1.0f);
    atomicAdd(&deg_out[src[i]], 1.0f);
  }
}

// p[i] = rsqrt(p[i] + 1)   (deg -> dinv with self-loop; deg+1 >= 1 always)
__global__ void rsqrtp1_kernel(float* __restrict__ p, long n) {
  long i = (long)blockIdx.x * blockDim.x + threadIdx.x;
  long stride = (long)gridDim.x * blockDim.x;
  for (; i < n; i += stride) p[i] = rsqrtf(p[i] + 1.0f);
}

// per-edge symmetric norms for both conv directions
__global__ void norm_kernel(const int* __restrict__ src, const int* __restrict__ dst,
                            const float* __restrict__ dinv_in, const float* __restrict__ dinv_out,
                            float* __restrict__ norm_in, float* __restrict__ norm_out, long E) {
  long i = (long)blockIdx.x * blockDim.x + threadIdx.x;
  long stride = (long)gridDim.x * blockDim.x;
  for (; i < E; i += stride) {
    int s = src[i], d = dst[i];
    norm_in[i]  = dinv_in[s]  * dinv_in[d];
    norm_out[i] = dinv_out[s] * dinv_out[d];
  }
}

// column sums / sumsq of x [N x 16] -> sums[0..16), sqs[0..16)
__global__ __launch_bounds__(256) void xstats_kernel(const float* __restrict__ x,
                                                     float* __restrict__ sums,
                                                     float* __restrict__ sqs, int NN) {
  __shared__ float ls[32];
  int tid = threadIdx.x;
  if (tid < 32) ls[tid] = 0.0f;
  __syncthreads();
  int col = tid & 15;  // stride is a multiple of 16 -> col fixed per thread
  float s = 0.0f, q = 0.0f;
  long total = (long)NN * 16;
  long stride = (long)gridDim.x * blockDim.x;
  for (long i = (long)blockIdx.x * blockDim.x + tid; i < total; i += stride) {
    float v = x[i];
    s += v; q += v * v;
  }
  atomicAdd(&ls[col], s);
  atomicAdd(&ls[16 + col], q);
  __syncthreads();
  if (tid < 16) atomicAdd(&sums[tid], ls[tid]);
  else if (tid < 32) atomicAdd(&sqs[tid - 16], ls[tid]);
}

// ------------------------------------------- BN1 fold: W' = a*W, b' = fc_b + shift.W
// also zeroes the second-BN accumulators for this layer
__global__ void fold_w_kernel(const float* __restrict__ sums, const float* __restrict__ sqs,
                              const float* __restrict__ gamma, const float* __restrict__ beta,
                              const float* __restrict__ W, const float* __restrict__ fb,
                              float* __restrict__ Wp, float* __restrict__ bp,
                              float* __restrict__ sum2, float* __restrict__ sq2,
                              int fcin, float invN) {
  __shared__ float a_s[64], sh_s[64];
  int tid = threadIdx.x;
  if (tid < fcin) {
    float m = sums[tid] * invN;
    float var = sqs[tid] * invN - m * m;
    float a = gamma[tid] * rsqrtf(var + BN_EPS);
    a_s[tid] = a;
    sh_s[tid] = beta[tid] - a * m;
  }
  if (tid < 32) { sum2[tid] = 0.0f; sq2[tid] = 0.0f; }
  __syncthreads();
  for (int idx = tid; idx < fcin * 32; idx += blockDim.x) {
    int c = idx >> 5;
    Wp[idx] = a_s[c] * W[idx];
  }
  if (tid < 32) {
    float acc = fb[tid];
    for (int c = 0; c < fcin; ++c) acc += sh_s[c] * W[c * 32 + tid];
    bp[tid] = acc;
  }
}

// ------------------------------------------- FC: h = relu(A @ W' + b') via f32 WMMA
// A rows come from [x | out], W' is [fcin x 32]. Fuses column sums/sumsq of h.
__global__ __launch_bounds__(256) void fc_wmma_kernel(
    const float* __restrict__ x,    // [N,16]
    const float* __restrict__ out,  // [N,64] (cols < 16*l valid)
    const float* __restrict__ Wp,   // [fcin,32]
    const float* __restrict__ bp,   // [32]
    float* __restrict__ h,          // [N,32]
    float* __restrict__ sum2, float* __restrict__ sq2,
    int NN, int fcin) {
  __shared__ float ls[64];  // 32 sums + 32 sumsq
  int tid = threadIdx.x;
  if (tid < 64) ls[tid] = 0.0f;
  __syncthreads();

  int wave = tid >> 5;
  int lane = tid & 31;
  int half = lane >> 4;   // 0/1
  int lidx = lane & 15;   // M for A, N for B/C
  int base = blockIdx.x * 128 + wave * 16;  // 16-row tile

  int arowi = base + lidx;
  bool rowok = arowi < NN;
  int arow = rowok ? arowi : (NN - 1);  // clamp address, zero value below

  v8f c0 = {};
  v8f c1 = {};
  for (int k0 = 0; k0 < fcin; k0 += 4) {
    int ka = k0 + 2 * half;
    int kb = ka + 1;
    float a0 = (ka < 16) ? x[arow * 16 + ka] : out[(long)arow * 64 + (ka - 16)];
    float a1 = (kb < 16) ? x[arow * 16 + kb] : out[(long)arow * 64 + (kb - 16)];
    if (!rowok) { a0 = 0.0f; a1 = 0.0f; }
    v2f A = {a0, a1};
    v2f B0 = {Wp[ka * 32 + lidx], Wp[kb * 32 + lidx]};
    v2f B1 = {Wp[ka * 32 + 16 + lidx], Wp[kb * 32 + 16 + lidx]};
    c0 = __builtin_amdgcn_wmma_f32_16x16x4_f32(false, A, false, B0, (short)0, c0, false, false);
    c1 = __builtin_amdgcn_wmma_f32_16x16x4_f32(false, A, false, B1, (short)0, c1, false, false);
  }

  float b0 = bp[lidx], b1 = bp[16 + lidx];
  float s0 = 0.0f, q0 = 0.0f, s1 = 0.0f, q1 = 0.0f;
#pragma unroll
  for (int p = 0; p < 8; ++p) {
    int M = half * 8 + p;          // C layout: VGPR p -> M = p + 8*half, N = lidx
    int row = base + M;
    float v0 = fmaxf(c0[p] + b0, 0.0f);
    float v1 = fmaxf(c1[p] + b1, 0.0f);
    if (row < NN) {
      h[(long)row * 32 + lidx] = v0;
      h[(long)row * 32 + 16 + lidx] = v1;
      s0 += v0; q0 += v0 * v0;
      s1 += v1; q1 += v1 * v1;
    }
  }
  atomicAdd(&ls[lidx], s0);
  atomicAdd(&ls[16 + lidx], s1);
  atomicAdd(&ls[32 + lidx], q0);
  atomicAdd(&ls[48 + lidx], q1);
  __syncthreads();
  if (tid < 32) atomicAdd(&sum2[tid], ls[tid]);
  else if (tid < 64) atomicAdd(&sq2[tid - 32], ls[tid]);
}

// ------------------------------------------- BN2 scale/shift
__global__ void bn2_prep_kernel(const float* __restrict__ sum2, const float* __restrict__ sq2,
                                const float* __restrict__ g, const float* __restrict__ b,
                                float* __restrict__ a2, float* __restrict__ b2, float invN) {
  int c = threadIdx.x;  // 32 threads
  float m = sum2[c] * invN;
  float var = sq2[c] * invN - m * m;
  float a = g[c] * rsqrtf(var + BN_EPS);
  a2[c] = a;
  b2[c] = b[c] - a * m;
}

// ------------------------------------------- z = relu(BN2(h)); t = z @ [W_in|W_out]
__global__ __launch_bounds__(256) void transform_kernel(
    const float* __restrict__ h, const float* __restrict__ a2, const float* __restrict__ b2,
    const float* __restrict__ Win, const float* __restrict__ Wout,
    float* __restrict__ t, int NN) {
  __shared__ float sa[32], sb[32], swi[256], swo[256];
  int tid = threadIdx.x;
  if (tid < 32) { sa[tid] = a2[tid]; sb[tid] = b2[tid]; }
  swi[tid] = Win[tid];   // 32*8 = 256
  swo[tid] = Wout[tid];
  __syncthreads();
  long stride = (long)gridDim.x * blockDim.x;
  for (long n = (long)blockIdx.x * blockDim.x + tid; n < NN; n += stride) {
    float z[32];
    const float4* hr = (const float4*)(h + n * 32);
#pragma unroll
    for (int c4 = 0; c4 < 8; ++c4) {
      float4 v = hr[c4];
      z[c4 * 4 + 0] = fmaxf(v.x * sa[c4 * 4 + 0] + sb[c4 * 4 + 0], 0.0f);
      z[c4 * 4 + 1] = fmaxf(v.y * sa[c4 * 4 + 1] + sb[c4 * 4 + 1], 0.0f);
      z[c4 * 4 + 2] = fmaxf(v.z * sa[c4 * 4 + 2] + sb[c4 * 4 + 2], 0.0f);
      z[c4 * 4 + 3] = fmaxf(v.w * sa[c4 * 4 + 3] + sb[c4 * 4 + 3], 0.0f);
    }
    float acc[16];
#pragma unroll
    for (int j = 0; j < 16; ++j) acc[j] = 0.0f;
#pragma unroll
    for (int c = 0; c < 32; ++c) {
      float zc = z[c];
#pragma unroll
      for (int j = 0; j < 8; ++j) {
        acc[j] += zc * swi[c * 8 + j];
        acc[8 + j] += zc * swo[c * 8 + j];
      }
    }
    float4* tw = (float4*)(t + n * 16);
    tw[0] = make_float4(acc[0], acc[1], acc[2], acc[3]);
    tw[1] = make_float4(acc[4], acc[5], acc[6], acc[7]);
    tw[2] = make_float4(acc[8], acc[9], acc[10], acc[11]);
    tw[3] = make_float4(acc[12], acc[13], acc[14], acc[15]);
  }
}

// ------------------------------------------- edge aggregation (both convs, one pass)
__global__ __launch_bounds__(256) void edge_agg_kernel(
    const int* __restrict__ src, const int* __restrict__ dst,
    const float* __restrict__ norm_in, const float* __restrict__ norm_out,
    const float* __restrict__ t, float* __restrict__ agg, long E) {
  long i = (long)blockIdx.x * blockDim.x + threadIdx.x;
  long stride = (long)gridDim.x * blockDim.x;
  for (; i < E; i += stride) {
    int s = src[i], d = dst[i];
    float ni = norm_in[i], no = norm_out[i];
    const float4* ts = (const float4*)(t + (long)s * 16);      // t_in  (cols 0-7)
    const float4* td = (const float4*)(t + (long)d * 16 + 8);  // t_out (cols 8-15)
    float4 ta = ts[0], tb = ts[1];
    float4 tc = td[0], te = td[1];
    float* ad = agg + (long)d * 16;      // in-conv accumulates at dst
    float* as = agg + (long)s * 16 + 8;  // out-conv accumulates at src
    atomicAdd(ad + 0, ni * ta.x); atomicAdd(ad + 1, ni * ta.y);
    atomicAdd(ad + 2, ni * ta.z); atomicAdd(ad + 3, ni * ta.w);
    atomicAdd(ad + 4, ni * tb.x); atomicAdd(ad + 5, ni * tb.y);
    atomicAdd(ad + 6, ni * tb.z); atomicAdd(ad + 7, ni * tb.w);
    atomicAdd(as + 0, no * tc.x); atomicAdd(as + 1, no * tc.y);
    atomicAdd(as + 2, no * tc.z); atomicAdd(as + 3, no * tc.w);
    atomicAdd(as + 4, no * te.x); atomicAdd(as + 5, no * te.y);
    atomicAdd(as + 6, no * te.z); atomicAdd(as + 7, no * te.w);
  }
}

// ------------------------------------------- finalize: + self loop + bias, write out,
// fuse column stats of the 16 new output columns (for next layer's BN1)
__global__ __launch_bounds__(256) void finalize_kernel(
    const float* __restrict__ agg, const float* __restrict__ t,
    const float* __restrict__ dinv_in, const float* __restrict__ dinv_out,
    const float* __restrict__ b_in, const float* __restrict__ b_out,
    float* __restrict__ out, float* __restrict__ sums, float* __restrict__ sqs,
    int NN, int layer, int do_stats) {
  __shared__ float ls[32];
  int tid = threadIdx.x;
  if (tid < 32) ls[tid] = 0.0f;
  __syncthreads();
  int j = tid & 15;  // stride multiple of 16 -> fixed column per thread
  float s = 0.0f, q = 0.0f;
  long total = (long)NN * 16;
  long stride = (long)gridDim.x * blockDim.x;
  for (long i = (long)blockIdx.x * blockDim.x + tid; i < total; i += stride) {
    long n = i >> 4;
    float dv = (j < 8) ? dinv_in[n] : dinv_out[n];
    float bias = (j < 8) ? b_in[j] : b_out[j - 8];
    float v = agg[i] + dv * dv * t[i] + bias;
    out[n * 64 + layer * 16 + j] = v;
    s += v; q += v * v;
  }
  if (do_stats) {
    atomicAdd(&ls[j], s);
    atomicAdd(&ls[16 + j], q);
    __syncthreads();
    int colbase = 16 + layer * 16;
    if (tid < 16) atomicAdd(&sums[colbase + tid], ls[tid]);
    else if (tid < 32) atomicAdd(&sqs[colbase + tid - 16], ls[tid]);
  }
}

// ---------------------------------------------------------------------------
extern "C" void kernel_launch(void* const* d_in, const int* in_sizes, int n_in,
                              void* d_out, int out_size, void* d_ws, size_t ws_size,
                              hipStream_t stream) {
  const float* x = (const float*)d_in[0];
  const int* ei = (const int*)d_in[1];
  const int* src = ei;
  const int* dst = ei + NEDGES;
  float* out = (float*)d_out;  // [N, 64]

  // params: per layer, JAX dict keys sorted:
  // 0 W_in, 1 W_out, 2 b_in, 3 b_out, 4 cv_beta, 5 cv_gamma, 6 fc_W, 7 fc_b, 8 fc_beta, 9 fc_gamma
  auto prm = [&](int l, int k) -> const float* { return (const float*)d_in[2 + 10 * l + k]; };

  // workspace layout (floats)
  float* ws = (float*)d_ws;
  float* dinv_in = ws;                      // N  (deg -> dinv in place)
  float* dinv_out = dinv_in + NNODES;       // N
  float* norm_in = dinv_out + NNODES;       // E
  float* norm_out = norm_in + NEDGES;       // E
  float* hbuf = norm_out + NEDGES;          // N*32
  float* tbuf = hbuf + (long)NNODES * 32;   // N*16
  float* agg = tbuf + (long)NNODES * 16;    // N*16
  float* sums = agg + (long)NNODES * 16;    // 80
  float* sqs = sums + 80;                   // 80
  float* sum2 = sqs + 80;                   // 32
  float* sq2 = sum2 + 32;                   // 32
  float* Wp = sq2 + 32;                     // 64*32
  float* bp = Wp + 64 * 32;                 // 32
  float* a2 = bp + 32;                      // 32
  float* b2 = a2 + 32;                      // 32

  const float invN = 1.0f / (float)NNODES;
  const int EB = (NEDGES + 255) / 256;      // 12500
  const int NB16 = ((long)NNODES * 16 + 255) / 256;

  // ---- once-per-call prep
  zero_f32_kernel<<<1, 256, 0, stream>>>(sums, 160);                    // sums+sqs
  zero_f32_kernel<<<1563, 256, 0, stream>>>(dinv_in, (long)2 * NNODES); // deg accumulators
  xstats_kernel<<<512, 256, 0, stream>>>(x, sums, sqs, NNODES);
  deg_kernel<<<EB, 256, 0, stream>>>(src, dst, dinv_in, dinv_out, NEDGES);
  rsqrtp1_kernel<<<1563, 256, 0, stream>>>(dinv_in, (long)2 * NNODES);
  norm_kernel<<<EB, 256, 0, stream>>>(src, dst, dinv_in, dinv_out, norm_in, norm_out, NEDGES);

  // ---- layers
  for (int l = 0; l < 4; ++l) {
    int fcin = 16 * (l + 1);
    fold_w_kernel<<<1, 256, 0, stream>>>(sums, sqs, prm(l, 9), prm(l, 8), prm(l, 6), prm(l, 7),
                                         Wp, bp, sum2, sq2, fcin, invN);
    fc_wmma_kernel<<<(NNODES + 127) / 128, 256, 0, stream>>>(x, out, Wp, bp, hbuf, sum2, sq2,
                                                             NNODES, fcin);
    bn2_prep_kernel<<<1, 32, 0, stream>>>(sum2, sq2, prm(l, 5), prm(l, 4), a2, b2, invN);
    transform_kernel<<<(NNODES + 255) / 256, 256, 0, stream>>>(hbuf, a2, b2, prm(l, 0), prm(l, 1),
                                                               tbuf, NNODES);
    zero_f32_kernel<<<NB16, 256, 0, stream>>>(agg, (long)NNODES * 16);
    edge_agg_kernel<<<EB, 256, 0, stream>>>(src, dst, norm_in, norm_out, tbuf, agg, NEDGES);
    finalize_kernel<<<NB16, 256, 0, stream>>>(agg, tbuf, dinv_in, dinv_out, prm(l, 2), prm(l, 3),
                                              out, sums, sqs, NNODES, l, (l < 3) ? 1 : 0);
  }
}